// SoftmaxCascade_48730698940767
// MI455X (gfx1250) — compile-verified
//
#include <hip/hip_runtime.h>

#define E_TOTAL 4681
#define NT      256
#define L2E     1.4426950408889634f

typedef __attribute__((address_space(1))) int* as1_i32p;
typedef __attribute__((address_space(3))) int* as3_i32p;

static __device__ __forceinline__ void async_copy_f32(const float* gsrc, float* ldst) {
#if __has_builtin(__builtin_amdgcn_global_load_async_to_lds_b32)
    __builtin_amdgcn_global_load_async_to_lds_b32((as1_i32p)gsrc, (as3_i32p)ldst, 0, 0);
#else
    *ldst = *gsrc;   // fallback: synchronous copy through VGPRs
#endif
}

static __device__ __forceinline__ void wait_async_zero() {
#if __has_builtin(__builtin_amdgcn_s_wait_asynccnt)
    __builtin_amdgcn_s_wait_asynccnt(0);
#else
    asm volatile("s_wait_asynccnt 0" ::: "memory");
#endif
}

static __device__ __forceinline__ float fast_exp2(float x) {
#if __has_builtin(__builtin_amdgcn_exp2f)
    return __builtin_amdgcn_exp2f(x);
#else
    return exp2f(x);
#endif
}

static __device__ __forceinline__ float fast_log2(float x) {
#if __has_builtin(__builtin_amdgcn_logf)
    return __builtin_amdgcn_logf(x);   // v_log_f32 is log base-2
#else
    return log2f(x);
#endif
}

__global__ __launch_bounds__(NT)
void SoftmaxCascade_48730698940767_kernel(const float* __restrict__ inputs,
                                          float* __restrict__ out) {
    __shared__ float sm[E_TOTAL];
    const int tid = threadIdx.x;
    const long long row = blockIdx.x;
    const float* __restrict__ src = inputs + row * (long long)E_TOTAL;
    float* __restrict__ dst = out + row * (long long)E_TOTAL;

    // ---- Stage the whole row into LDS via the async global->LDS pipe ----
    for (int j = tid; j < E_TOTAL; j += NT)
        async_copy_f32(src + j, &sm[j]);
    wait_async_zero();     // each wave drains its own ASYNCcnt
    __syncthreads();       // then all waves' data is visible

    // Tree layout (branch=8, depth=4): levels start at {0,1,9,73,585}
    const int childBase[5]  = {0, 1, 9, 73, 585};
    const int parentBase[5] = {0, 0, 1, 9,  73};
    const int ngroups[5]    = {0, 1, 8, 64, 512};

#pragma unroll
    for (int l = 1; l <= 4; ++l) {
        for (int g = tid; g < ngroups[l]; g += NT) {
            // parent log-prob (log2 domain); root conditional is identically 0
            const float p = (l == 1) ? 0.0f : sm[parentBase[l] + g];
            const int b = childBase[l] + (g << 3);

            float x[8];
#pragma unroll
            for (int k = 0; k < 8; ++k) x[k] = sm[b + k];

            float m = x[0];
#pragma unroll
            for (int k = 1; k < 8; ++k) m = fmaxf(m, x[k]);

            float s = 0.0f;
#pragma unroll
            for (int k = 0; k < 8; ++k) s += fast_exp2((x[k] - m) * L2E);

            // stored logp (base-2) = (x-m)*L2E - log2(s) + parent
            const float c = p - fmaf(m, L2E, fast_log2(s));
#pragma unroll
            for (int k = 0; k < 8; ++k) sm[b + k] = fmaf(x[k], L2E, c);
        }
        if (l == 1 && tid == 0) sm[0] = 0.0f;  // root: log p = 0
        __syncthreads();
    }

    // ---- exp and coalesced non-temporal store of all 4681 nodes ----
    for (int j = tid; j < E_TOTAL; j += NT) {
        const float v = fast_exp2(sm[j]);
        __builtin_nontemporal_store(v, dst + j);
    }
}

extern "C" void kernel_launch(void* const* d_in, const int* in_sizes, int n_in,
                              void* d_out, int out_size, void* d_ws, size_t ws_size,
                              hipStream_t stream) {
    const float* inputs = (const float*)d_in[0];   // [B, E] float32
    // d_in[1] = segment_ids, d_in[2] = path_onehot: structure is static, folded
    // into the kernel's hardcoded tree recurrence.
    float* out = (float*)d_out;                    // [B, E] float32
    const int batch = in_sizes[0] / E_TOTAL;
    hipLaunchKernelGGL(SoftmaxCascade_48730698940767_kernel,
                       dim3(batch), dim3(NT), 0, stream, inputs, out);
}